// ConvolutionalCapsule_47021301956943
// MI455X (gfx1250) — compile-verified
//
#include <hip/hip_runtime.h>
#include <hip/hip_bf16.h>

// ---------------------------------------------------------------------------
// Convolutional capsule layer, 2 routing iterations, all-f32 on CDNA5 WMMA.
//   pass1: centroids1 = (1/32) * sum_c preds  -> squash -> out1   (WMMA GEMM)
//   pass2: logits = <preds, out1>_o                                (WMMA + shfl)
//   pass3: softmax over F axis (reference softmaxes axis=3!)
//   pass4: centroids2 = sum_c cc*preds -> squash -> d_out          (WMMA GEMM)
// Workspace: 32.5 MB (Apack 5.3 + Wt 4.7 + out1 1.2 + cc 21.2), L2-resident.
// ---------------------------------------------------------------------------

typedef float v2f __attribute__((ext_vector_type(2)));
typedef float v8f __attribute__((ext_vector_type(8)));

#define NB    4
#define HIN   14
#define FIN   32
#define DI    8
#define HO    12
#define NF    32
#define NC    288          // 3*3*32 capsules per patch
#define DO_   16
#define NM    (NB*HO*HO)   // 576 spatial positions
#define NT    (NM/16)      // 36 M-tiles
#define EPS_  1e-7f

// workspace layout (float offsets)
#define SZ_APACK (NT*NC*128)         // 1,327,104  packed A fragments
#define SZ_WT    (NF*NC*128)         // 1,179,648  packed B fragments
#define SZ_OUT1  (NM*NF*DO_)         //   294,912
#define SZ_CC    (NT*NF*NC*16)       // 5,308,416  logits -> cc (in place)
#define OFF_APACK 0
#define OFF_WT    (OFF_APACK + SZ_APACK)
#define OFF_OUT1  (OFF_WT + SZ_WT)
#define OFF_CC    (OFF_OUT1 + SZ_OUT1)

// Fragment convention for V_WMMA_F32_16X16X4_F32 (wave32):
//   A (16x4): lane L, reg v  -> A[M = L%16, K = v + 2*(L/16)]
//   B (4x16): lane L, reg v  -> B[K = v + 2*(L/16), N = L%16]
//   C/D     : lane L, reg j  -> D[M = j + 8*(L/16), N = L%16]

// ---- pack patches -> A fragments: Apack[tile][c][half][lane][v] ------------
__global__ void pack_patches(const float* __restrict__ x, float* __restrict__ ap) {
    int t = blockIdx.x * blockDim.x + threadIdx.x;
    if (t >= SZ_APACK) return;
    int v    = t & 1;
    int L    = (t >> 1) & 31;
    int half = (t >> 6) & 1;
    int c    = (t >> 7) % NC;
    int tile = t / (NC * 128);
    int m   = tile * 16 + (L & 15);
    int b   = m / (HO * HO);
    int rem = m % (HO * HO);
    int h   = rem / HO, w = rem % HO;
    int kh  = c / (3 * FIN);
    int kw  = (c / FIN) % 3;
    int fin = c % FIN;
    int i   = v + 2 * (L >> 4) + 4 * half;
    ap[t] = x[(((b * HIN + h + kh) * HIN + (w + kw)) * FIN + fin) * DI + i];
}

// ---- pack W -> B fragments: Wt[f][c][half][lane][v] ------------------------
__global__ void pack_weights(const float* __restrict__ Wsrc, float* __restrict__ wt) {
    int t = blockIdx.x * blockDim.x + threadIdx.x;
    if (t >= SZ_WT) return;
    int v    = t & 1;
    int L    = (t >> 1) & 31;
    int half = (t >> 6) & 1;
    int c    = (t >> 7) % NC;
    int f    = t / (NC * 128);
    int o    = L & 15;
    int i    = v + 2 * (L >> 4) + 4 * half;
    wt[t] = Wsrc[((f * NC + c) * DO_ + o) * DI + i];
}

__device__ __forceinline__ v8f wmma_f32(v2f a, v2f b, v8f c) {
    return __builtin_amdgcn_wmma_f32_16x16x4_f32(
        /*neg_a=*/false, a, /*neg_b=*/false, b,
        /*c_mod=*/(short)0, c, /*reuse_a=*/false, /*reuse_b=*/false);
}

// ---- GEMM + squash (pass1: cc=1/32 uniform; pass4: cc from routing) --------
template <bool PASS2>
__global__ __launch_bounds__(32) void capsule_pass(
    const float* __restrict__ apack, const float* __restrict__ wt,
    const float* __restrict__ ccT, float* __restrict__ dst) {
    const int tile = blockIdx.x;
    const int f    = blockIdx.y;
    const int L    = threadIdx.x;

    const v2f* A2 = (const v2f*)apack;
    const v2f* B2 = (const v2f*)wt;
    const int abase  = tile * NC * 64 + L;      // 64 float2 per c (2 halves x 32 lanes)
    const int bbase  = f * NC * 64 + L;
    const int ccbase = ((f * NT + tile) * NC) * 16 + (L & 15);

    v8f acc = {0.f, 0.f, 0.f, 0.f, 0.f, 0.f, 0.f, 0.f};
    for (int c = 0; c < NC; ++c) {
        v2f a0 = A2[abase + c * 64];
        v2f a1 = A2[abase + c * 64 + 32];
        v2f b0 = B2[bbase + c * 64];
        v2f b1 = B2[bbase + c * 64 + 32];
        if (PASS2) {
            float cv = ccT[ccbase + c * 16];    // cc[m = L%16, c] (softmaxed)
            a0[0] *= cv; a0[1] *= cv;
            a1[0] *= cv; a1[1] *= cv;
        }
        acc = wmma_f32(a0, b0, acc);
        acc = wmma_f32(a1, b1, acc);
    }
    // squash each row m over o (16 lanes within the wave32 half)
    const int mbase = tile * 16 + ((L >> 4) << 3);
    const int o     = L & 15;
#pragma unroll
    for (int j = 0; j < 8; ++j) {
        float s = acc[j];
        if (!PASS2) s *= (1.0f / 32.0f);
        float sn = s * s;
        sn += __shfl_xor(sn, 1, 32);
        sn += __shfl_xor(sn, 2, 32);
        sn += __shfl_xor(sn, 4, 32);
        sn += __shfl_xor(sn, 8, 32);
        float scale = (sn / (1.0f + sn)) / sqrtf(sn + EPS_);
        dst[(size_t)(mbase + j) * (NF * DO_) + f * DO_ + o] = s * scale;
    }
}

// ---- agreement: logits[m,f,c] = sum_o preds[m,f,c,o]*out1[m,f,o] -----------
__global__ __launch_bounds__(32) void agreement_kernel(
    const float* __restrict__ apack, const float* __restrict__ wt,
    const float* __restrict__ out1, float* __restrict__ ccT) {
    const int tile = blockIdx.x;
    const int f    = blockIdx.y;
    const int L    = threadIdx.x;

    const v2f* A2 = (const v2f*)apack;
    const v2f* B2 = (const v2f*)wt;
    const int abase = tile * NC * 64 + L;
    const int bbase = f * NC * 64 + L;
    const int o     = L & 15;

    float o1[8];
#pragma unroll
    for (int j = 0; j < 8; ++j) {  // out1 in the C/D register layout
        int m = tile * 16 + j + ((L >> 4) << 3);
        o1[j] = out1[(size_t)m * (NF * DO_) + f * DO_ + o];
    }
    const int outbase = ((f * NT + tile) * NC) * 16;

    for (int c = 0; c < NC; ++c) {
        v2f a0 = A2[abase + c * 64];
        v2f a1 = A2[abase + c * 64 + 32];
        v2f b0 = B2[bbase + c * 64];
        v2f b1 = B2[bbase + c * 64 + 32];
        v8f p = {0.f, 0.f, 0.f, 0.f, 0.f, 0.f, 0.f, 0.f};
        p = wmma_f32(a0, b0, p);       // preds tile [16m x 16o] for this c
        p = wmma_f32(a1, b1, p);
        float vals[8];
#pragma unroll
        for (int j = 0; j < 8; ++j) {  // dot over o across the 16-lane group
            float t2 = p[j] * o1[j];
            t2 += __shfl_xor(t2, 1, 32);
            t2 += __shfl_xor(t2, 2, 32);
            t2 += __shfl_xor(t2, 4, 32);
            t2 += __shfl_xor(t2, 8, 32);
            vals[j] = t2;
        }
        if (L == 0) {
#pragma unroll
            for (int j = 0; j < 8; ++j) ccT[outbase + c * 16 + j] = vals[j];
        } else if (L == 16) {
#pragma unroll
            for (int j = 0; j < 8; ++j) ccT[outbase + c * 16 + 8 + j] = vals[j];
        }
    }
}

// ---- softmax over F axis (axis=3 in the reference), in place ---------------
__global__ void softmax_f(float* __restrict__ ccT) {
    const int n = NT * NC * 16;  // items per f-plane; f-stride == n
    int t = blockIdx.x * blockDim.x + threadIdx.x;
    if (t >= n) return;
    float v[NF];
    float mx = -1e30f;
#pragma unroll
    for (int f = 0; f < NF; ++f) {
        v[f] = ccT[(size_t)f * n + t];
        mx = fmaxf(mx, v[f]);
    }
    float s = 0.f;
#pragma unroll
    for (int f = 0; f < NF; ++f) {
        v[f] = expf(v[f] - mx);
        s += v[f];
    }
    float inv = 1.0f / s;
#pragma unroll
    for (int f = 0; f < NF; ++f) ccT[(size_t)f * n + t] = v[f] * inv;
}

extern "C" void kernel_launch(void* const* d_in, const int* in_sizes, int n_in,
                              void* d_out, int out_size, void* d_ws, size_t ws_size,
                              hipStream_t stream) {
    const float* x = (const float*)d_in[0];   // (4,14,14,32,8)
    const float* W = (const float*)d_in[1];   // (1,1,1,32,288,16,8)
    float* ws   = (float*)d_ws;               // needs 32,440,320 bytes
    float* ap   = ws + OFF_APACK;
    float* wt   = ws + OFF_WT;
    float* out1 = ws + OFF_OUT1;
    float* cc   = ws + OFF_CC;

    pack_patches<<<(SZ_APACK + 255) / 256, 256, 0, stream>>>(x, ap);
    pack_weights<<<(SZ_WT + 255) / 256, 256, 0, stream>>>(W, wt);

    dim3 grid(NT, NF);
    capsule_pass<false><<<grid, 32, 0, stream>>>(ap, wt, nullptr, out1);
    agreement_kernel<<<grid, 32, 0, stream>>>(ap, wt, out1, cc);
    softmax_f<<<(NT * NC * 16 + 255) / 256, 256, 0, stream>>>(cc);
    capsule_pass<true><<<grid, 32, 0, stream>>>(ap, wt, cc, (float*)d_out);
}